// SparseArch_38482906972957
// MI455X (gfx1250) — compile-verified
//
#include <hip/hip_runtime.h>

#define ZCH   400000
#define NIDS  1000000
#define EMBD  128

typedef __attribute__((ext_vector_type(2))) float v2f;
typedef __attribute__((ext_vector_type(8))) float v8f;

// ---------------------------------------------------------------------------
// Kernel A: MCH remap (lower-bound binary search) + slot histogram.
// Writes remapped index (as float) into d_out[1 + feat*N + i].
// Overflow slot (ZCH-1) hits ~90% of ids -> aggregate per-wave via ballot
// so we issue ~1 atomic per wave instead of ~1 per lane on that address.
// ---------------------------------------------------------------------------
__global__ void remap_hist_kernel(const int* __restrict__ ids0,
                                  const int* __restrict__ ids1,
                                  const int* __restrict__ mch0,
                                  const int* __restrict__ mch1,
                                  unsigned* __restrict__ hist0,
                                  unsigned* __restrict__ hist1,
                                  float* __restrict__ out_remap) {
  const int feat = blockIdx.y;
  const int* __restrict__ ids = feat ? ids1 : ids0;
  const int* __restrict__ mch = feat ? mch1 : mch0;
  unsigned* __restrict__ hist = feat ? hist1 : hist0;
  float* __restrict__ outp = out_remap + (size_t)feat * NIDS;

  int i = blockIdx.x * blockDim.x + threadIdx.x;
  if (i >= NIDS) return;

  const int id = ids[i];
  int lo = 0, hi = ZCH;
  // lower_bound (searchsorted side='left'), 19 iterations; table is L2-resident
  while (lo < hi) {
    int mid = (lo + hi) >> 1;
    if (mch[mid] < id) lo = mid + 1; else hi = mid;
  }
  int pos = (lo < (ZCH - 1)) ? lo : (ZCH - 1);
  bool hit = (mch[pos] == id);
  int r = hit ? pos : (ZCH - 1);
  outp[i] = (float)r;

  bool last = (r == ZCH - 1);
  if (!last) {
    atomicAdd(&hist[r], 1u);  // low contention: spread over 400K slots
  }
  // wave-aggregated update of the hot overflow slot
  unsigned long long mlast = __ballot(last);
  int lane = threadIdx.x & 31;
  if (last && lane == (__ffsll(mlast) - 1)) {
    atomicAdd(&hist[ZCH - 1], (unsigned)__popcll(mlast));
  }
}

// ---------------------------------------------------------------------------
// Kernel B: loss partial = sum_s count[s] * rowsum(emb[s]) via WMMA.
// Each wave handles 16-row tiles. A = count-weighted emb values (f32, exact
// since counts < 2^24), B = ones (D[m][n] = rowsum(m) for every column n).
// A-layout (f32 16x4): lanes 0-15 hold K=0,1 ; lanes 16-31 hold K=2,3.
// Streams each 205MB table exactly once -> ~18us at 23.3 TB/s.
// ---------------------------------------------------------------------------
__global__ void wsum_wmma_kernel(const float* __restrict__ emb0,
                                 const float* __restrict__ emb1,
                                 const unsigned* __restrict__ hist0,
                                 const unsigned* __restrict__ hist1,
                                 double* __restrict__ acc) {
  const int feat = blockIdx.y;
  const float* __restrict__ emb = feat ? emb1 : emb0;
  const unsigned* __restrict__ hist = feat ? hist1 : hist0;

  const int lane  = threadIdx.x & 31;
  const int wave  = (blockIdx.x * blockDim.x + threadIdx.x) >> 5;
  const int nwave = (gridDim.x * blockDim.x) >> 5;
  const int ntiles = ZCH / 16;  // 25000, exact

  v8f c = {};          // 16x16 f32 accumulator, 8 VGPRs
  v2f b;               // B = ones: layout-irrelevant
  b[0] = 1.0f; b[1] = 1.0f;

  const int col = (lane < 16) ? 0 : 2;  // K-pair this lane supplies

  for (int tile = wave; tile < ntiles; tile += nwave) {
    const int row = tile * 16 + (lane & 15);
    const float w = (float)hist[row];  // exact uint->f32 (counts < 2^24)
    const float* __restrict__ rp = emb + (size_t)row * EMBD + col;

    if (tile + nwave < ntiles) {  // wave-uniform: stream-ahead hint
      const float* np = emb + ((size_t)(tile + nwave) * 16 + (lane & 15)) * EMBD;
      __builtin_prefetch(np, 0, 1);
    }

#pragma unroll 8
    for (int kc = 0; kc < 32; ++kc) {  // 32 chunks of K=4 cover 128 dims
      v2f a;
      a[0] = rp[kc * 4]     * w;
      a[1] = rp[kc * 4 + 1] * w;
      // D = A x ones + C  => every column of D accumulates the row sums
      c = __builtin_amdgcn_wmma_f32_16x16x4_f32(
          /*neg_a=*/false, a, /*neg_b=*/false, b,
          /*c_mod=*/(short)0, c, /*reuse_a=*/false, /*reuse_b=*/false);
    }
  }

  // Column n of D lives in lane n (M=0..7) and lane n+16 (M=8..15);
  // all columns are identical, so lane L + lane (L^16) hold the full 16 rows.
  float csum = c[0] + c[1] + c[2] + c[3] + c[4] + c[5] + c[6] + c[7];
  float other = __shfl(csum, lane ^ 16, 32);
  float tot = csum + other;
  if (lane == 0) {
    atomicAdd(acc, (double)tot);  // ~8K f64 atomics total: negligible
  }
}

// ---------------------------------------------------------------------------
// Kernel C: finalize loss = acc / (2*N*D)
// ---------------------------------------------------------------------------
__global__ void finalize_kernel(const double* __restrict__ acc,
                                float* __restrict__ out_loss) {
  if (threadIdx.x == 0 && blockIdx.x == 0) {
    out_loss[0] = (float)(acc[0] / (double)((size_t)2 * NIDS * EMBD));
  }
}

extern "C" void kernel_launch(void* const* d_in, const int* in_sizes, int n_in,
                              void* d_out, int out_size, void* d_ws, size_t ws_size,
                              hipStream_t stream) {
  const int*   ids0 = (const int*)d_in[0];
  const int*   ids1 = (const int*)d_in[1];
  const int*   mch0 = (const int*)d_in[2];
  const int*   mch1 = (const int*)d_in[3];
  const float* emb0 = (const float*)d_in[4];
  const float* emb1 = (const float*)d_in[5];
  float* out = (float*)d_out;

  // workspace layout: hist0 [ZCH u32] | hist1 [ZCH u32] | acc [1 f64]
  unsigned* hist0 = (unsigned*)d_ws;
  unsigned* hist1 = hist0 + ZCH;
  double*   acc   = (double*)(hist1 + ZCH);  // 8B-aligned (offset 3.2MB)

  hipMemsetAsync(d_ws, 0, (size_t)2 * ZCH * sizeof(unsigned) + sizeof(double),
                 stream);

  dim3 gA((NIDS + 255) / 256, 2);
  remap_hist_kernel<<<gA, 256, 0, stream>>>(ids0, ids1, mch0, mch1,
                                            hist0, hist1, out + 1);

  dim3 gB(512, 2);  // 4096 waves/feature, ~6 tiles each
  wsum_wmma_kernel<<<gB, 256, 0, stream>>>(emb0, emb1, hist0, hist1, acc);

  finalize_kernel<<<1, 1, 0, stream>>>(acc, out);
}